// RBF_1125281431629
// MI455X (gfx1250) — compile-verified
//
#include <hip/hip_runtime.h>
#include <stdint.h>

// RBF activation: out[n,c] = sum_j w[c,j] * exp(-0.5*((x[n,c]-mu_j)/sigma)^2)
// x: [8,128,128,64] f32 (NHWC, C=64 contiguous), w: [64,31] f32.
// K=31 centers on linspace(-0.25,0.25), sigma = 0.5/30.
//
// Strategy (MI455X / gfx1250, wave32):
//  * trans-pipe bound op: 31 v_exp_f32 per element; 3 VALU + 1 TRANS per center
//    by working in the exp2 domain: e^{-d^2/2} = 2^{-(k*d)^2}, k=sqrt(log2e/2).
//  * lane pinned to a single channel (all element strides % 64 == 0) -> the 31
//    per-channel weights live in VGPRs; no steady-state LDS/global weight reads.
//  * w (64x31 f32 = 7936B) staged block-wide into LDS once via the CDNA5
//    Tensor Data Mover (tensor_load_to_lds), fenced with s_wait_tensorcnt.
//  * 4 independent element chains (TRANS/VALU co-execution) x 4 outer iters
//    = 16 elements/thread to amortize weight-load + setup overhead.
//  * non-temporal streaming loads/stores (x/out each touched exactly once).

#define FILTERS 64
#define KW 31
#define NITER 4            // outer iterations per thread (4 elements each)

typedef uint32_t u32;
typedef uint64_t u64;
typedef __attribute__((ext_vector_type(4))) u32 v4u;
typedef __attribute__((ext_vector_type(8))) int v8i;
typedef __attribute__((ext_vector_type(4))) int v4i;

__global__ __launch_bounds__(256) void rbf_kernel(const float* __restrict__ x,
                                                  const float* __restrict__ w,
                                                  float* __restrict__ out,
                                                  int n) {
    __shared__ float lw[FILTERS * KW];  // 7936 bytes

    // ---- Stage w into LDS with the Tensor Data Mover (one wave issues) ----
    if (threadIdx.x < 32) {
        const u32 nelem = FILTERS * KW;              // 1984 elements, 4B each
        u32 lds_off = (u32)(uintptr_t)&lw[0];        // flat LDS addr: low 32 bits = offset
        u64 ga = (u64)(uintptr_t)w;

        // D# group 0: count=1 | lds_addr | global_addr[56:0] | type=2
        v4u g0 = (v4u){ 0x1u,
                        lds_off,
                        (u32)ga,
                        (u32)((ga >> 32) & 0x01FFFFFFu) | 0x80000000u };
        // D# group 1: data_size=4B; 1-D tensor: dim0=1984, dim1=1,
        //             tile_dim0=1984, tile_dim1=1, dim0_stride=1984
        v8i g1 = (v8i){ (int)(2u << 16),                              // data_size=2 (4B)
                        (int)((nelem & 0xFFFFu) << 16),               // tensor_dim0 lo16 @bit48
                        (int)(((nelem >> 16) & 0xFFFFu) | (1u << 16)),// dim0 hi | tensor_dim1=1
                        (int)(nelem << 16),                           // tile_dim0 @bits127:112
                        (int)1,                                       // tile_dim1 = 1
                        (int)nelem,                                   // tensor_dim0_stride lo32
                        0, 0 };
        v4i g2 = (v4i){0, 0, 0, 0};
        v4i g3 = (v4i){0, 0, 0, 0};
        v8i g4 = (v8i){0, 0, 0, 0, 0, 0, 0, 0};      // 6-arg form: extra group (unused)
        __builtin_amdgcn_tensor_load_to_lds(g0, g1, g2, g3, g4, 0);
        __builtin_amdgcn_s_wait_tensorcnt(0);
    }
    __syncthreads();

    // ---- Per-lane channel is fixed: c = tid % 64 (grid*block % 64 == 0) ----
    const int c = (int)(threadIdx.x & (FILTERS - 1));
    float wr[KW];
#pragma unroll
    for (int j = 0; j < KW; ++j) wr[j] = lw[c * KW + j];  // 31-stride: 64-bank conflict-free

    // u_j = k*(x - VMIN)/sigma - j*k ;  gauss_j = 2^(-u_j^2) = e^{-d_j^2/2}
    constexpr float KS = 0.84932214f;           // sqrt(0.5*log2(e))
    constexpr float A  = 60.0f * KS;            // ((K-1)/(vmax-vmin)) * KS
    constexpr float B  = 15.0f * KS;            // (-vmin)*60 * KS

    const u64 T   = (u64)gridDim.x * blockDim.x;
    const u64 tid = (u64)blockIdx.x * blockDim.x + threadIdx.x;
    const u64 nn  = (u64)n;

#pragma unroll 1
    for (int it = 0; it < NITER; ++it) {
        float u[4], acc[4];
#pragma unroll
        for (int q = 0; q < 4; ++q) {
            u64 e = tid + (u64)(4 * it + q) * T;   // strides mult of 64: channel fixed
            float xv = (e < nn) ? __builtin_nontemporal_load(&x[e]) : 0.0f;
            u[q]   = fmaf(xv, A, B);
            acc[q] = 0.0f;
        }

#pragma unroll
        for (int j = 0; j < KW; ++j) {
#pragma unroll
            for (int q = 0; q < 4; ++q) {
                float g = __builtin_amdgcn_exp2f(-(u[q] * u[q]));  // v_exp_f32
                acc[q] = fmaf(wr[j], g, acc[q]);
                u[q] -= KS;
            }
        }

#pragma unroll
        for (int q = 0; q < 4; ++q) {
            u64 e = tid + (u64)(4 * it + q) * T;
            if (e < nn) __builtin_nontemporal_store(acc[q], &out[e]);
        }
    }
}

extern "C" void kernel_launch(void* const* d_in, const int* in_sizes, int n_in,
                              void* d_out, int out_size, void* d_ws, size_t ws_size,
                              hipStream_t stream) {
    const float* x = (const float*)d_in[0];   // [8*128*128*64] f32
    const float* w = (const float*)d_in[1];   // [64*31] f32
    float* out = (float*)d_out;
    int n = in_sizes[0];                       // 8,388,608

    // 16 elements per thread; 256 threads/block (8 wave32 waves).
    const int per_block = 256 * 4 * NITER;     // 4096
    int blocks = (int)(((u64)n + per_block - 1) / per_block);
    rbf_kernel<<<blocks, 256, 0, stream>>>(x, w, out, n);
}